// HGCN_53017076302391
// MI455X (gfx1250) — compile-verified
//
#include <hip/hip_runtime.h>
#include <hip/hip_bf16.h>
#include <math.h>

// ---------------------------------------------------------------------------
// HGCN forward (2 layers), MI455X / gfx1250, wave32.
// Precision: f32 end-to-end (matches reference) using V_WMMA_F32_16X16X4_F32
// for the mobius_matvec GEMM. Edge aggregation uses hardware
// global_atomic_add_f32 (inline asm).
// ---------------------------------------------------------------------------

#define DD 256
#define MINN 1e-15f
#define MAXN 0.996f            // 1 - PROJ_EPS, c = 1
#define CLIP 0.99999988f       // ~ 1 - 1e-7 representable in f32

typedef __attribute__((ext_vector_type(2))) float v2f;
typedef __attribute__((ext_vector_type(8))) float v8f;

__device__ __forceinline__ float artanh_c(float x) {
    x = fminf(fmaxf(x, -CLIP), CLIP);
    return atanhf(x);
}

// Block-wide sum over blockDim.x = 256 threads (8 wave32 waves).
__device__ __forceinline__ float block_reduce_sum(float v, float* sbuf) {
#pragma unroll
    for (int off = 16; off > 0; off >>= 1)
        v += __shfl_xor(v, off, 32);
    const int lane = threadIdx.x & 31;
    const int wid  = threadIdx.x >> 5;
    __syncthreads();                  // protect sbuf across repeated calls
    if (lane == 0) sbuf[wid] = v;
    __syncthreads();
    float s = 0.0f;
    const int nw = blockDim.x >> 5;
#pragma unroll
    for (int i = 0; i < 8; ++i)
        if (i < nw) s += sbuf[i];
    return s;
}

// Hardware f32 atomic add (no-return) — guaranteed global_atomic_add_f32.
__device__ __forceinline__ void atomic_add_f32(float* p, float v) {
    asm volatile("global_atomic_add_f32 %0, %1, off" :: "v"(p), "v"(v) : "memory");
}

// ---------------------------------------------------------------------------
// GEMM: MX = H(NxD) * W^T  (W is DxD row-major; MX[n][j] = sum_k H[n][k]W[j][k])
// Block: 512 thr = 16 waves. Block tile: 16 rows x 256 cols; wave tile 16x16.
// ---------------------------------------------------------------------------
#define HS_STRIDE 258   // pad: even (8B ds loads) and co-prime-ish with 64 banks

__global__ void __launch_bounds__(512)
hgcn_gemm_wmma(const float* __restrict__ H, const float* __restrict__ W,
               float* __restrict__ MX, int N) {
    __shared__ float hs[16 * HS_STRIDE];
    const int i0 = blockIdx.x * 16;

    // cooperatively stage the 16x256 H tile
    for (int idx = threadIdx.x; idx < 16 * DD; idx += blockDim.x) {
        const int r = idx >> 8, c = idx & 255;
        const int gr = i0 + r;
        hs[r * HS_STRIDE + c] = (gr < N) ? H[gr * DD + c] : 0.0f;
    }
    __syncthreads();

    const int wave = threadIdx.x >> 5;   // 0..15 -> column tile
    const int lane = threadIdx.x & 31;
    const int lm   = lane & 15;
    const int hi   = lane >> 4;          // K sub-offset select
    const int j0   = wave * 16;

    const float* wrow = W + (j0 + lm) * DD;   // row (j0+lm) of W
    const float* arow = hs + lm * HS_STRIDE;

    v8f acc = {};
    for (int k = 0; k < DD; k += 4) {
        const int ka = k + hi * 2;
        const v2f a = { arow[ka], arow[ka + 1] };        // ds_load_b64
        const v2f b = { wrow[ka], wrow[ka + 1] };        // global_load_b64
        acc = __builtin_amdgcn_wmma_f32_16x16x4_f32(
                  false, a, false, b, (short)0, acc, false, false);
    }

    const int col   = j0 + lm;
    const int rbase = i0 + hi * 8;
#pragma unroll
    for (int r = 0; r < 8; ++r) {
        const int row = rbase + r;
        if (row < N) MX[row * DD + col] = acc[r];
    }
}

// ---------------------------------------------------------------------------
// h = proj(expmap0(x)) : combined scale = min(tanh(|u|), MAXN)/|u|
// ---------------------------------------------------------------------------
__global__ void __launch_bounds__(256)
hgcn_expmap_proj(const float* __restrict__ X, float* __restrict__ O) {
    __shared__ float sbuf[8];
    const int row = blockIdx.x;
    const float v = X[row * DD + threadIdx.x];
    const float s = block_reduce_sum(v * v, sbuf);
    const float un = fmaxf(sqrtf(s), MINN);
    const float sc = fminf(tanhf(un), MAXN) / un;
    O[row * DD + threadIdx.x] = sc * v;
}

// hyp_bias = proj(expmap0(b)); store vector [0..255] and ||.||^2 at [256]
__global__ void __launch_bounds__(256)
hgcn_bias(const float* __restrict__ B, float* __restrict__ bias) {
    __shared__ float sbuf[8];
    const float v = B[threadIdx.x];
    const float s = block_reduce_sum(v * v, sbuf);
    const float un = fmaxf(sqrtf(s), MINN);
    const float sc = fminf(tanhf(un), MAXN) / un;
    const float hb = sc * v;
    bias[threadIdx.x] = hb;
    const float s2 = block_reduce_sum(hb * hb, sbuf);
    if (threadIdx.x == 0) bias[DD] = s2;
}

// ---------------------------------------------------------------------------
// HypLinear tail + logmap0:
//  res = proj(mobius_matvec scale * mx); h = proj(mobius_add(res,bias));
//  xt = logmap0(h).  In-place safe when XT aliases MX (reads happen first).
// ---------------------------------------------------------------------------
__global__ void __launch_bounds__(256)
hgcn_hyplinear_logmap(const float* __restrict__ H, const float* __restrict__ MX,
                      const float* __restrict__ bias, float* __restrict__ XT) {
    __shared__ float sbuf[8];
    const int row = blockIdx.x;
    const int d   = threadIdx.x;
    const float hv = H[row * DD + d];
    const float mv = MX[row * DD + d];

    const float sh = block_reduce_sum(hv * hv, sbuf);
    const float sm = block_reduce_sum(mv * mv, sbuf);
    const float xn = fmaxf(sqrtf(sh), MINN);
    const float mn = fmaxf(sqrtf(sm), MINN);

    float r = tanhf((mn / xn) * artanh_c(xn)) / mn;   // mobius_matvec scale
    if (sm == 0.0f) r = 0.0f;                         // zero_row guard
    float res = r * mv;
    float rn  = fabsf(r) * sqrtf(sm);                 // ||res||
    const float p1 = (rn > MAXN) ? (MAXN / fmaxf(rn, MINN)) : 1.0f;
    res *= p1;
    const float rn2 = fminf(rn, MAXN);
    const float x2  = rn2 * rn2;

    const float bv = bias[d];
    const float y2 = bias[DD];
    const float xy = block_reduce_sum(res * bv, sbuf);

    const float den = fmaxf(1.0f + 2.0f * xy + x2 * y2, MINN);
    float hd = ((1.0f + 2.0f * xy + y2) * res + (1.0f - x2) * bv) / den;

    const float s3 = block_reduce_sum(hd * hd, sbuf);
    const float hn = sqrtf(s3);
    const float p2 = (hn > MAXN) ? (MAXN / fmaxf(hn, MINN)) : 1.0f;
    hd *= p2;
    const float pn = fmaxf(fminf(hn, MAXN), MINN);
    const float lg = artanh_c(pn) / pn;               // logmap0 scale
    XT[row * DD + d] = lg * hd;
}

// ---------------------------------------------------------------------------
__global__ void __launch_bounds__(256)
hgcn_zero(float* __restrict__ p, int n) {
    for (int i = blockIdx.x * blockDim.x + threadIdx.x; i < n;
         i += gridDim.x * blockDim.x)
        p[i] = 0.0f;
}

// one block per edge: agg[dst] += w * xt[src]
__global__ void __launch_bounds__(256)
hgcn_edge(const float* __restrict__ XT, const float* __restrict__ EW,
          const int* __restrict__ ES, const int* __restrict__ ED,
          float* __restrict__ AG, int E) {
    const int e = blockIdx.x;
    if (e >= E) return;
    const int   s = ES[e];
    const int   t = ED[e];
    const float w = EW[e];
    const float v = w * XT[s * DD + threadIdx.x];
    atomic_add_f32(&AG[t * DD + threadIdx.x], v);
}

// ---------------------------------------------------------------------------
// h = proj(expmap0(agg)); xt = relu(logmap0(h)); out = proj(expmap0(xt))
// ---------------------------------------------------------------------------
__global__ void __launch_bounds__(256)
hgcn_agg_act(const float* __restrict__ AG, float* __restrict__ O) {
    __shared__ float sbuf[8];
    const int row = blockIdx.x;
    const int d   = threadIdx.x;
    const float a = AG[row * DD + d];

    const float s  = block_reduce_sum(a * a, sbuf);
    const float un = fmaxf(sqrtf(s), MINN);
    const float t  = fminf(tanhf(un), MAXN);
    const float hd = (t / un) * a;                    // proj(expmap0(agg))
    const float pn = fmaxf(t, MINN);                  // ||h||
    const float lg = artanh_c(pn) / pn;               // logmap0 scale
    const float v  = fmaxf(lg * hd, 0.0f);            // relu in tangent space

    const float s2 = block_reduce_sum(v * v, sbuf);
    const float vn = fmaxf(sqrtf(s2), MINN);
    const float sc = fminf(tanhf(vn), MAXN) / vn;
    O[row * DD + d] = sc * v;
}

// ---------------------------------------------------------------------------
extern "C" void kernel_launch(void* const* d_in, const int* in_sizes, int n_in,
                              void* d_out, int out_size, void* d_ws, size_t ws_size,
                              hipStream_t stream) {
    const float* x  = (const float*)d_in[0];
    const float* W1 = (const float*)d_in[1];
    const float* b1 = (const float*)d_in[2];
    const float* W2 = (const float*)d_in[3];
    const float* b2 = (const float*)d_in[4];
    const float* ew = (const float*)d_in[5];
    const int*   es = (const int*)d_in[6];
    const int*   ed = (const int*)d_in[7];

    const int N = in_sizes[0] / DD;   // 10000
    const int E = in_sizes[5];        // 320000

    float* ws   = (float*)d_ws;
    float* h    = ws;                      // [N*DD]
    float* mxt  = h   + (size_t)N * DD;    // [N*DD]  mx, then reused as xt
    float* agg  = mxt + (size_t)N * DD;    // [N*DD]
    float* bias = agg + (size_t)N * DD;    // [DD+1]

    const int gemmGrid = (N + 15) / 16;
    const int nElems   = N * DD;

    // h0 = proj(expmap0(x))
    hgcn_expmap_proj<<<N, 256, 0, stream>>>(x, h);

    const float* Ws[2] = { W1, W2 };
    const float* Bs[2] = { b1, b2 };
    for (int layer = 0; layer < 2; ++layer) {
        float* out = (layer == 1) ? (float*)d_out : h;
        hgcn_gemm_wmma<<<gemmGrid, 512, 0, stream>>>(h, Ws[layer], mxt, N);
        hgcn_bias<<<1, 256, 0, stream>>>(Bs[layer], bias);
        hgcn_hyplinear_logmap<<<N, 256, 0, stream>>>(h, mxt, bias, mxt);
        hgcn_zero<<<2048, 256, 0, stream>>>(agg, nElems);
        hgcn_edge<<<E, 256, 0, stream>>>(mxt, ew, es, ed, agg, E);
        hgcn_agg_act<<<N, 256, 0, stream>>>(agg, out);
    }
}